// GATConv_4363686772847
// MI455X (gfx1250) — compile-verified
//
#include <hip/hip_runtime.h>
#include <math.h>

typedef float v2f __attribute__((ext_vector_type(2)));
typedef float v8f __attribute__((ext_vector_type(8)));

#define DIM 128
#define NEG_SLOPE 0.2f
#define GAT_EPS 1e-16f

// ---------------- K0: init out / amax / denom ----------------
__global__ void k_init(float* __restrict__ out, float* __restrict__ amax,
                       float* __restrict__ denom, int N) {
    int i = blockIdx.x * blockDim.x + threadIdx.x;
    int total = N * DIM;
    if (i < total) out[i] = 0.0f;
    if (i < N) {
        amax[i]  = -__builtin_inff();
        denom[i] = 0.0f;
    }
}

// ---------------- K1: h = x @ W via V_WMMA_F32_16X16X4_F32 ----------------
// block = 256 threads = 8 waves. blockIdx.x -> 16-row tile; wave id -> 16-col tile.
__global__ void k_gemm_wmma(const float* __restrict__ x, const float* __restrict__ W,
                            float* __restrict__ h, int N) {
    const int wave = threadIdx.x >> 5;
    const int lane = threadIdx.x & 31;
    const int row0 = blockIdx.x * 16;
    const int col0 = wave * 16;

    const int m  = lane & 15;          // row within tile (A), col within tile (B)
    const int kh = (lane >> 4) * 2;    // K sub-offset: lanes 0-15 -> K+{0,1}, 16-31 -> K+{2,3}

    int arow = row0 + m;
    if (arow >= N) arow = N - 1;       // clamp: keep EXEC all-ones through WMMA
    const float* __restrict__ xrow = x + (size_t)arow * DIM;

    v8f acc = {};
    for (int k0 = 0; k0 < DIM; k0 += 4) {
        v2f a, b;
        // A 16x4 layout: lane m holds row m; VGPR0/1 = K = kh / kh+1
        a.x = xrow[k0 + kh];
        a.y = xrow[k0 + kh + 1];
        // B 4x16 layout: lane m holds col m; VGPR0/1 = K = kh / kh+1
        b.x = W[(size_t)(k0 + kh)     * DIM + col0 + m];
        b.y = W[(size_t)(k0 + kh + 1) * DIM + col0 + m];
        acc = __builtin_amdgcn_wmma_f32_16x16x4_f32(
            /*neg_a=*/false, a, /*neg_b=*/false, b,
            /*c_mod=*/(short)0, acc, /*reuse_a=*/false, /*reuse_b=*/false);
    }

    // C/D layout: VGPR v, lanes 0-15 -> M=v, lanes 16-31 -> M=v+8, N=lane&15
    const int rbase = row0 + (lane >> 4) * 8;
    const int c = col0 + m;
#pragma unroll
    for (int v = 0; v < 8; ++v) {
        int r = rbase + v;
        if (r < N) h[(size_t)r * DIM + c] = acc[v];
    }
}

// ---------------- K2: per-node scores a_src = h.att_src, a_dst = h.att_dst ----
__global__ void k_scores(const float* __restrict__ h, const float* __restrict__ att_src,
                         const float* __restrict__ att_dst,
                         float* __restrict__ a_src, float* __restrict__ a_dst, int N) {
    const int gtid = blockIdx.x * blockDim.x + threadIdx.x;
    const int node = gtid >> 5;
    const int lane = threadIdx.x & 31;
    if (node >= N) return;
    const float* __restrict__ hr = h + (size_t)node * DIM;
    float s = 0.0f, d = 0.0f;
#pragma unroll
    for (int j = lane; j < DIM; j += 32) {
        float v = hr[j];
        s += v * att_src[j];
        d += v * att_dst[j];
    }
#pragma unroll
    for (int off = 16; off > 0; off >>= 1) {
        s += __shfl_xor(s, off, 32);
        d += __shfl_xor(d, off, 32);
    }
    if (lane == 0) { a_src[node] = s; a_dst[node] = d; }
}

// monotone float atomic max: signed-int max for >=0, unsigned-int min for <0
__device__ __forceinline__ void atomicMaxF(float* addr, float val) {
    if (val >= 0.0f) atomicMax((int*)addr, __float_as_int(val));
    else             atomicMin((unsigned int*)addr, __float_as_uint(val));
}

// ---------------- K3: per-edge leaky-relu score + segment max ----------------
__global__ void k_edge_score(const int* __restrict__ ei,
                             const float* __restrict__ a_src, const float* __restrict__ a_dst,
                             float* __restrict__ alpha, float* __restrict__ amax,
                             long long E, int N) {
    long long e = (long long)blockIdx.x * blockDim.x + threadIdx.x;
    long long Etot = E + (long long)N;
    if (e >= Etot) return;
    int s, d;
    if (e < E) { s = ei[e]; d = ei[E + e]; }
    else       { s = d = (int)(e - E); }       // self loop
    float a = a_src[s] + a_dst[d];
    a = (a > 0.0f) ? a : NEG_SLOPE * a;        // leaky relu
    alpha[e] = a;
    atomicMaxF(&amax[d], a);
}

// ---------------- K4: per-edge exp + segment sum ----------------
__global__ void k_edge_exp(const int* __restrict__ ei,
                           float* __restrict__ alpha, const float* __restrict__ amax,
                           float* __restrict__ denom, long long E, int N) {
    long long e = (long long)blockIdx.x * blockDim.x + threadIdx.x;
    long long Etot = E + (long long)N;
    if (e >= Etot) return;
    int d;
    if (e < E) d = ei[E + e];
    else       d = (int)(e - E);
    float ex = expf(alpha[e] - amax[d]);
    alpha[e] = ex;
    atomicAdd(&denom[d], ex);
}

// ---------------- K5: wave-per-edge weighted scatter-add ----------------
__global__ void k_edge_aggr(const int* __restrict__ ei,
                            const float* __restrict__ alpha, const float* __restrict__ denom,
                            const float* __restrict__ h, float* __restrict__ out,
                            long long E, int N) {
    long long gtid = (long long)blockIdx.x * blockDim.x + threadIdx.x;
    long long e = gtid >> 5;
    const int lane = threadIdx.x & 31;
    long long Etot = E + (long long)N;
    if (e >= Etot) return;
    int s, d;
    if (e < E) { s = ei[e]; d = ei[E + e]; }
    else       { s = d = (int)(e - E); }
    const float coef = alpha[e] / (denom[d] + GAT_EPS);
    const float* __restrict__ hs = h + (size_t)s * DIM;
    float* __restrict__ od = out + (size_t)d * DIM;
#pragma unroll
    for (int j = lane; j < DIM; j += 32) {
        atomicAdd(&od[j], hs[j] * coef);   // global_atomic_add_f32, coalesced per wave
    }
}

extern "C" void kernel_launch(void* const* d_in, const int* in_sizes, int n_in,
                              void* d_out, int out_size, void* d_ws, size_t ws_size,
                              hipStream_t stream) {
    const float* x       = (const float*)d_in[0];
    const int*   ei      = (const int*)d_in[1];   // edge_index [2, E] int32
    const float* W       = (const float*)d_in[2];
    const float* att_src = (const float*)d_in[3];
    const float* att_dst = (const float*)d_in[4];

    const int       N = in_sizes[0] / DIM;
    const long long E = (long long)in_sizes[1] / 2;
    const long long Etot = E + (long long)N;

    float* out = (float*)d_out;

    // workspace layout
    float* h      = (float*)d_ws;              // N*DIM
    float* a_src  = h + (size_t)N * DIM;       // N
    float* a_dst  = a_src + N;                 // N
    float* amax   = a_dst + N;                 // N
    float* denom  = amax + N;                  // N
    float* alpha  = denom + N;                 // E + N

    // K0: init
    {
        long long tot = (long long)N * DIM;
        int blocks = (int)((tot + 255) / 256);
        k_init<<<blocks, 256, 0, stream>>>(out, amax, denom, N);
    }
    // K1: projection via WMMA (8 waves/block = 8 column tiles of DIM=128)
    {
        int mtiles = (N + 15) / 16;
        k_gemm_wmma<<<mtiles, 256, 0, stream>>>(x, W, h, N);
    }
    // K2: per-node scores, one wave per node
    {
        long long threads = (long long)N * 32;
        int blocks = (int)((threads + 255) / 256);
        k_scores<<<blocks, 256, 0, stream>>>(h, att_src, att_dst, a_src, a_dst, N);
    }
    // K3: edge scores + segment max
    {
        int blocks = (int)((Etot + 255) / 256);
        k_edge_score<<<blocks, 256, 0, stream>>>(ei, a_src, a_dst, alpha, amax, E, N);
    }
    // K4: exp + segment sum
    {
        int blocks = (int)((Etot + 255) / 256);
        k_edge_exp<<<blocks, 256, 0, stream>>>(ei, alpha, amax, denom, E, N);
    }
    // K5: wave-per-edge weighted aggregation
    {
        long long threads = Etot * 32;
        int blocks = (int)((threads + 255) / 256);
        k_edge_aggr<<<blocks, 256, 0, stream>>>(ei, alpha, denom, h, out, E, N);
    }
}